// GlobalFreqBranch_128849019511
// MI455X (gfx1250) — compile-verified
//
#include <hip/hip_runtime.h>
#include <math.h>

// ---------------------------------------------------------------------------
// GlobalFreqBranch on MI455X (gfx1250, wave32)
// DFT-as-GEMM via V_WMMA_F32_16X16X4_F32; MLP second GEMM also on WMMA f32.
// ---------------------------------------------------------------------------

typedef float v2f __attribute__((ext_vector_type(2)));
typedef float v8f __attribute__((ext_vector_type(8)));

#define IMGN   224
#define NPIX   (IMGN * IMGN)     // 50176
#define BATCH  64
#define NBAND  8
#define EDIM   256

static __device__ __forceinline__ v8f wmma_f32(v2f a, v2f b, v8f c) {
  // (neg_a, A, neg_b, B, c_mod, C, reuse_a, reuse_b)
  return __builtin_amdgcn_wmma_f32_16x16x4_f32(false, a, false, b, (short)0, c,
                                               false, false);
}

// ---------------------------------------------------------------------------
// K0: DFT matrices  Fr[n][k] = cos(2*pi*n*k/N), Fi[n][k] = -sin(2*pi*n*k/N)
// ---------------------------------------------------------------------------
__global__ void k_init(float* __restrict__ Fr, float* __restrict__ Fi) {
  int idx = blockIdx.x * blockDim.x + threadIdx.x;
  if (idx >= NPIX) return;
  int n = idx / IMGN, k = idx % IMGN;
  int m = (n * k) % IMGN;                       // exact in int
  float theta = -6.283185307179586f * (float)m / (float)IMGN;
  Fr[idx] = cosf(theta);
  Fi[idx] = sinf(theta);
}

// ---------------------------------------------------------------------------
// K1: grayscale = channel mean
// ---------------------------------------------------------------------------
__global__ void k_gray(const float* __restrict__ x, float* __restrict__ g) {
  int i = blockIdx.x * blockDim.x + threadIdx.x;   // over BATCH*NPIX
  if (i >= BATCH * NPIX) return;
  int b = i / NPIX, p = i % NPIX;
  const float* xb = x + (size_t)b * 3 * NPIX;
  g[i] = (xb[p] + xb[NPIX + p] + xb[2 * NPIX + p]) * (1.0f / 3.0f);
}

// ---------------------------------------------------------------------------
// K2: stage 1 of 2D DFT:  Yr = g*Fr, Yi = g*Fi   (per image 224x224 GEMMs)
// one 16x16 output tile per wave, 14x14=196 tiles/image, 8 waves/block
// ---------------------------------------------------------------------------
__global__ void k_dft1(const float* __restrict__ g,
                       const float* __restrict__ Fr,
                       const float* __restrict__ Fi,
                       float* __restrict__ Yr, float* __restrict__ Yi) {
  int b = blockIdx.y;
  int wave = threadIdx.x >> 5;
  int tile = blockIdx.x * 8 + wave;
  if (tile >= 196) return;                       // wave-uniform exit
  int tm = tile / 14, tn = tile % 14;
  int lane = threadIdx.x & 31;
  int m = lane & 15;                             // A row / B col / D col
  int half = lane >> 4;                          // selects K pair / D row-half
  const float* gb = g + (size_t)b * NPIX;
  int arow = tm * 16 + m;
  int bcol = tn * 16 + m;

  v8f accr = {};
  v8f acci = {};
  for (int k0 = 0; k0 < IMGN; k0 += 4) {
    int ka = k0 + half * 2;
    v2f a;  a.x  = gb[arow * IMGN + ka];       a.y  = gb[arow * IMGN + ka + 1];
    v2f br; br.x = Fr[ka * IMGN + bcol];       br.y = Fr[(ka + 1) * IMGN + bcol];
    v2f bi; bi.x = Fi[ka * IMGN + bcol];       bi.y = Fi[(ka + 1) * IMGN + bcol];
    accr = wmma_f32(a, br, accr);
    acci = wmma_f32(a, bi, acci);
  }
  float* yr = Yr + (size_t)b * NPIX;
  float* yi = Yi + (size_t)b * NPIX;
#pragma unroll
  for (int v = 0; v < 8; ++v) {
    int orow = tm * 16 + v + half * 8;
    int ocol = tn * 16 + m;
    yr[orow * IMGN + ocol] = accr[v];
    yi[orow * IMGN + ocol] = acci[v];
  }
}

// ---------------------------------------------------------------------------
// K3: stage 2:  Gr = Fr*Yr - Fi*Yi ; Gi = Fr*Yi + Fi*Yr
// fused epilogue: mag = log1p(|G|), stored fftshifted (+112 mod 224)
// ---------------------------------------------------------------------------
__global__ void k_dft2(const float* __restrict__ Fr,
                       const float* __restrict__ Fi,
                       const float* __restrict__ Yr,
                       const float* __restrict__ Yi,
                       float* __restrict__ mag) {
  int b = blockIdx.y;
  int wave = threadIdx.x >> 5;
  int tile = blockIdx.x * 8 + wave;
  if (tile >= 196) return;
  int tm = tile / 14, tn = tile % 14;
  int lane = threadIdx.x & 31;
  int m = lane & 15;
  int half = lane >> 4;
  const float* yr = Yr + (size_t)b * NPIX;
  const float* yi = Yi + (size_t)b * NPIX;
  int arow = tm * 16 + m;
  int bcol = tn * 16 + m;

  v8f a_rr = {};   // Fr*Yr
  v8f a_ii = {};   // Fi*Yi
  v8f a_ri = {};   // Fr*Yi
  v8f a_ir = {};   // Fi*Yr
  for (int k0 = 0; k0 < IMGN; k0 += 4) {
    int ka = k0 + half * 2;
    v2f ar; ar.x = Fr[arow * IMGN + ka];   ar.y = Fr[arow * IMGN + ka + 1];
    v2f ai; ai.x = Fi[arow * IMGN + ka];   ai.y = Fi[arow * IMGN + ka + 1];
    v2f br; br.x = yr[ka * IMGN + bcol];   br.y = yr[(ka + 1) * IMGN + bcol];
    v2f bi; bi.x = yi[ka * IMGN + bcol];   bi.y = yi[(ka + 1) * IMGN + bcol];
    a_rr = wmma_f32(ar, br, a_rr);
    a_ii = wmma_f32(ai, bi, a_ii);
    a_ri = wmma_f32(ar, bi, a_ri);
    a_ir = wmma_f32(ai, br, a_ir);
  }
  float* mb = mag + (size_t)b * NPIX;
#pragma unroll
  for (int v = 0; v < 8; ++v) {
    int orow = tm * 16 + v + half * 8;
    int ocol = tn * 16 + m;
    float gr = a_rr[v] - a_ii[v];
    float gi = a_ri[v] + a_ir[v];
    float mgv = log1pf(sqrtf(gr * gr + gi * gi));
    int sr = (orow + 112) % IMGN;               // fftshift
    int sc = (ocol + 112) % IMGN;
    mb[sr * IMGN + sc] = mgv;
  }
}

// ---------------------------------------------------------------------------
// K4: per-image mean + unbiased std of mag
// ---------------------------------------------------------------------------
__global__ void k_stats(const float* __restrict__ mag, float* __restrict__ stats) {
  int b = blockIdx.x;
  int t = threadIdx.x;
  const float* mb = mag + (size_t)b * NPIX;
  float s = 0.0f, s2 = 0.0f;
  for (int p = t; p < NPIX; p += 256) {
    float v = mb[p];
    s += v; s2 += v * v;
  }
  for (int off = 16; off; off >>= 1) {
    s  += __shfl_down(s, off, 32);
    s2 += __shfl_down(s2, off, 32);
  }
  __shared__ float red[2][8];
  int wave = t >> 5, lane = t & 31;
  if (lane == 0) { red[0][wave] = s; red[1][wave] = s2; }
  __syncthreads();
  if (t == 0) {
    float ts = 0.0f, ts2 = 0.0f;
    for (int w = 0; w < 8; ++w) { ts += red[0][w]; ts2 += red[1][w]; }
    float mu  = ts / (float)NPIX;
    float var = (ts2 - (float)NPIX * mu * mu) / (float)(NPIX - 1);
    stats[b * 2 + 0] = mu;
    stats[b * 2 + 1] = sqrtf(fmaxf(var, 0.0f)) + 1e-6f;
  }
}

// ---------------------------------------------------------------------------
// K5: per-image radial-band stats -> feats[b][band][{mean,std,max}]
// ---------------------------------------------------------------------------
__global__ void k_band(const float* __restrict__ mag,
                       const float* __restrict__ stats,
                       float* __restrict__ feats) {
  int b = blockIdx.x;
  int t = threadIdx.x;
  float mu = stats[b * 2 + 0];
  float sd = stats[b * 2 + 1];
  const float step = (float)(224.0 / 1.4142135623730951 / 8.0);
  const float* mb = mag + (size_t)b * NPIX;

  float ls1[NBAND], ls2[NBAND], lmx[NBAND], lcn[NBAND];
#pragma unroll
  for (int k = 0; k < NBAND; ++k) { ls1[k] = 0; ls2[k] = 0; lcn[k] = 0; lmx[k] = -1e9f; }

  for (int p = t; p < NPIX; p += 256) {
    int y = p / IMGN, x = p % IMGN;
    float dx = (float)(x - 112), dy = (float)(y - 112);
    float r = sqrtf(dx * dx + dy * dy);
    int band = (int)(r / step);
    if (band < NBAND) {
      float mn = (mb[p] - mu) / sd;
      ls1[band] += mn;
      ls2[band] += mn * mn;
      lcn[band] += 1.0f;
      lmx[band] = fmaxf(lmx[band], mn);
    }
  }
  __shared__ float red[4][8][NBAND];   // [stat][wave][band]
  int wave = t >> 5, lane = t & 31;
#pragma unroll
  for (int k = 0; k < NBAND; ++k) {
    float a = ls1[k], c = ls2[k], n = lcn[k], mx = lmx[k];
    for (int off = 16; off; off >>= 1) {
      a  += __shfl_down(a, off, 32);
      c  += __shfl_down(c, off, 32);
      n  += __shfl_down(n, off, 32);
      mx  = fmaxf(mx, __shfl_down(mx, off, 32));
    }
    if (lane == 0) { red[0][wave][k] = a; red[1][wave][k] = c; red[2][wave][k] = n; red[3][wave][k] = mx; }
  }
  __syncthreads();
  if (t < NBAND) {
    float s1 = 0, s2 = 0, cn = 0, mx = -1e9f;
    for (int w = 0; w < 8; ++w) {
      s1 += red[0][w][t]; s2 += red[1][w][t]; cn += red[2][w][t];
      mx = fmaxf(mx, red[3][w][t]);
    }
    float denom = cn + 1e-6f;
    float mean  = s1 / denom;
    float varn  = s2 - 2.0f * mean * s1 + mean * mean * cn;
    float stdv  = sqrtf(varn / denom + 1e-6f);
    float* f = feats + ((size_t)b * NBAND + t) * 3;
    f[0] = mean; f[1] = stdv; f[2] = mx;
  }
}

// ---------------------------------------------------------------------------
// K6: MLP.  feats[512x3] @ W1[3x256] + b1 -> LN -> ReLU -> @ W2[256x256] + b2
// one block per 16 output rows; 8 waves x 2 N-tiles WMMA f32
// ---------------------------------------------------------------------------
__global__ void k_mlp(const float* __restrict__ feats,
                      const float* __restrict__ W1, const float* __restrict__ b1,
                      const float* __restrict__ gamma, const float* __restrict__ beta,
                      const float* __restrict__ W2, const float* __restrict__ b2,
                      float* __restrict__ out) {
  __shared__ float h[16][260];           // padded pitch to spread banks
  __shared__ float smean[16], srstd[16];
  int blk = blockIdx.x;                  // 0..31, rows blk*16..blk*16+15
  int t = threadIdx.x;                   // 256 threads

  // Phase 1: h = feats @ W1 + b1
  for (int i = 0; i < 16; ++i) {
    int grow = blk * 16 + i;
    float f0 = feats[grow * 3 + 0];
    float f1 = feats[grow * 3 + 1];
    float f2 = feats[grow * 3 + 2];
    h[i][t] = f0 * W1[t] + f1 * W1[EDIM + t] + f2 * W1[2 * EDIM + t] + b1[t];
  }
  __syncthreads();

  // Phase 2: LayerNorm statistics (one row per thread, tiny)
  if (t < 16) {
    float s = 0.0f, s2 = 0.0f;
    for (int e = 0; e < EDIM; ++e) { float v = h[t][e]; s += v; s2 += v * v; }
    float mean = s / (float)EDIM;
    float var  = s2 / (float)EDIM - mean * mean;
    smean[t] = mean;
    srstd[t] = rsqrtf(var + 1e-5f);
  }
  __syncthreads();
  for (int i = 0; i < 16; ++i) {
    float v = (h[i][t] - smean[i]) * srstd[i] * gamma[t] + beta[t];
    h[i][t] = fmaxf(v, 0.0f);
  }
  __syncthreads();

  // Phase 3: out = h @ W2 + b2 with WMMA f32 16x16x4
  int wave = t >> 5, lane = t & 31;
  int m = lane & 15, half = lane >> 4;
  for (int tt = 0; tt < 2; ++tt) {
    int tn = wave * 2 + tt;              // N-tile 0..15
    v8f acc = {};
    for (int k0 = 0; k0 < EDIM; k0 += 4) {
      int ka = k0 + half * 2;
      v2f a;  a.x  = h[m][ka];                       a.y  = h[m][ka + 1];
      v2f bb; bb.x = W2[ka * EDIM + tn * 16 + m];    bb.y = W2[(ka + 1) * EDIM + tn * 16 + m];
      acc = wmma_f32(a, bb, acc);
    }
#pragma unroll
    for (int v = 0; v < 8; ++v) {
      int orow = blk * 16 + v + half * 8;
      int oc   = tn * 16 + m;
      out[(size_t)orow * EDIM + oc] = acc[v] + b2[oc];
    }
  }
}

// ---------------------------------------------------------------------------
// Host launcher
// ---------------------------------------------------------------------------
extern "C" void kernel_launch(void* const* d_in, const int* in_sizes, int n_in,
                              void* d_out, int out_size, void* d_ws, size_t ws_size,
                              hipStream_t stream) {
  const float* x     = (const float*)d_in[0];
  const float* W1    = (const float*)d_in[1];
  const float* b1    = (const float*)d_in[2];
  const float* gamma = (const float*)d_in[3];
  const float* beta  = (const float*)d_in[4];
  const float* W2    = (const float*)d_in[5];
  const float* b2    = (const float*)d_in[6];
  float* out = (float*)d_out;

  char* ws = (char*)d_ws;
  const size_t szF   = (size_t)NPIX * sizeof(float);          // 200704 B
  const size_t szImg = (size_t)BATCH * NPIX * sizeof(float);  // 12.8 MB
  float* Fr    = (float*)(ws);
  float* Fi    = (float*)(ws + szF);
  float* bufA  = (float*)(ws + 2 * szF);                      // gray, then mag
  float* Yr    = (float*)(ws + 2 * szF + szImg);
  float* Yi    = (float*)(ws + 2 * szF + 2 * szImg);
  float* stats = (float*)(ws + 2 * szF + 3 * szImg);
  float* feats = (float*)(ws + 2 * szF + 3 * szImg + 1024);

  k_init<<<dim3((NPIX + 255) / 256), dim3(256), 0, stream>>>(Fr, Fi);
  k_gray<<<dim3((BATCH * NPIX + 255) / 256), dim3(256), 0, stream>>>(x, bufA);
  k_dft1<<<dim3(25, BATCH), dim3(256), 0, stream>>>(bufA, Fr, Fi, Yr, Yi);
  k_dft2<<<dim3(25, BATCH), dim3(256), 0, stream>>>(Fr, Fi, Yr, Yi, bufA);
  k_stats<<<dim3(BATCH), dim3(256), 0, stream>>>(bufA, stats);
  k_band<<<dim3(BATCH), dim3(256), 0, stream>>>(bufA, stats, feats);
  k_mlp<<<dim3(32), dim3(256), 0, stream>>>(feats, W1, b1, gamma, beta, W2, b2, out);
}